// cfrm_imp_88356067213424
// MI455X (gfx1250) — compile-verified
//
#include <hip/hip_runtime.h>
#include <hip/hip_bf16.h>

typedef __attribute__((ext_vector_type(2))) float v2f;
typedef __attribute__((ext_vector_type(8))) float v8f;

#define TPB 256

__device__ __forceinline__ void softmax3(float a, float b, float c,
                                         float& e0, float& e1, float& e2) {
  float m = fmaxf(a, fmaxf(b, c));
  e0 = expf(a - m); e1 = expf(b - m); e2 = expf(c - m);
  float inv = 1.0f / (e0 + e1 + e2);
  e0 *= inv; e1 *= inv; e2 *= inv;
}

// ---------------------------------------------------------------------------
// K_init: zero the scalar accumulators
// ---------------------------------------------------------------------------
__global__ void init_kernel(float* __restrict__ acc) {
  if (threadIdx.x < 4) acc[threadIdx.x] = 0.0f;
}

// ---------------------------------------------------------------------------
// K0: all softmaxes + tiny 3x3 matmuls (P1@A_g, P2@A_r) + row norms for MMD
// ---------------------------------------------------------------------------
__global__ void prep_kernel(const float* __restrict__ C1, const float* __restrict__ C2,
                            const float* __restrict__ Cr,
                            const float* __restrict__ Ag, const float* __restrict__ Ar,
                            const int* __restrict__ i1, const int* __restrict__ i2,
                            const int* __restrict__ ir,
                            int rc, int m1, int m2, int mr,
                            float* __restrict__ S0, float* __restrict__ S1, float* __restrict__ S2,
                            float* __restrict__ Prp,
                            float* __restrict__ P1p, float* __restrict__ Q1p, float* __restrict__ nsq1,
                            float* __restrict__ P2p, float* __restrict__ Q2p, float* __restrict__ nsq2)
{
  int t = blockIdx.x * blockDim.x + threadIdx.x;
  if (t < rc) {                                   // S = softmax(C_r), SoA layout
    float e0, e1, e2;
    softmax3(Cr[3*t], Cr[3*t+1], Cr[3*t+2], e0, e1, e2);
    S0[t] = e0; S1[t] = e1; S2[t] = e2;
  }
  if (t < mr) {                                   // Pr = softmax(C_r[mask_r]), padded [mr,4]
    int row = ir[t];
    float e0, e1, e2;
    softmax3(Cr[3*row], Cr[3*row+1], Cr[3*row+2], e0, e1, e2);
    Prp[4*t] = e0; Prp[4*t+1] = e1; Prp[4*t+2] = e2; Prp[4*t+3] = 0.0f;
  }
  if (t < m1) {                                   // P1, Q1 = P1@A_g, |P1|^2
    int row = i1[t];
    float p0, p1, p2;
    softmax3(C1[3*row], C1[3*row+1], C1[3*row+2], p0, p1, p2);
    P1p[4*t] = p0; P1p[4*t+1] = p1; P1p[4*t+2] = p2; P1p[4*t+3] = 0.0f;
    nsq1[t] = p0*p0 + p1*p1 + p2*p2;
    Q1p[4*t+0] = p0*Ag[0] + p1*Ag[3] + p2*Ag[6];
    Q1p[4*t+1] = p0*Ag[1] + p1*Ag[4] + p2*Ag[7];
    Q1p[4*t+2] = p0*Ag[2] + p1*Ag[5] + p2*Ag[8];
    Q1p[4*t+3] = 0.0f;
  }
  if (t < m2) {                                   // P2, Q2 = P2@A_r, |P2|^2
    int row = i2[t];
    float p0, p1, p2;
    softmax3(C2[3*row], C2[3*row+1], C2[3*row+2], p0, p1, p2);
    P2p[4*t] = p0; P2p[4*t+1] = p1; P2p[4*t+2] = p2; P2p[4*t+3] = 0.0f;
    nsq2[t] = p0*p0 + p1*p1 + p2*p2;
    Q2p[4*t+0] = p0*Ar[0] + p1*Ar[3] + p2*Ar[6];
    Q2p[4*t+1] = p0*Ar[1] + p1*Ar[4] + p2*Ar[7];
    Q2p[4*t+2] = p0*Ar[2] + p1*Ar[5] + p2*Ar[8];
    Q2p[4*t+3] = 0.0f;
  }
}

// ---------------------------------------------------------------------------
// K1: V = A[mask_g] @ S   (gather commuted through row-wise matmul: 200 MB
// instead of 800 MB). One block per output row; float4-coalesced streaming.
// ---------------------------------------------------------------------------
__global__ void vrow_kernel(const float* __restrict__ A, const int* __restrict__ ig,
                            const float* __restrict__ S0, const float* __restrict__ S1,
                            const float* __restrict__ S2,
                            int rc, int mg, float* __restrict__ Vp)
{
  __shared__ float r0[TPB], r1[TPB], r2[TPB];
  int b = blockIdx.x;
  size_t row = (size_t)ig[b];
  const float* arow = A + row * (size_t)rc;
  int n4 = rc >> 2;
  float s0 = 0.f, s1 = 0.f, s2 = 0.f;
  for (int k = threadIdx.x; k < n4; k += blockDim.x) {
    float4 a  = *(const float4*)(arow + 4*k);
    float4 t0 = *(const float4*)(S0 + 4*k);
    float4 t1 = *(const float4*)(S1 + 4*k);
    float4 t2 = *(const float4*)(S2 + 4*k);
    s0 += a.x*t0.x + a.y*t0.y + a.z*t0.z + a.w*t0.w;
    s1 += a.x*t1.x + a.y*t1.y + a.z*t1.z + a.w*t1.w;
    s2 += a.x*t2.x + a.y*t2.y + a.z*t2.z + a.w*t2.w;
  }
  if (threadIdx.x < (rc & 3)) {                   // generic tail
    int k = (n4 << 2) + threadIdx.x;
    float a = arow[k];
    s0 += a * S0[k]; s1 += a * S1[k]; s2 += a * S2[k];
  }
  r0[threadIdx.x] = s0; r1[threadIdx.x] = s1; r2[threadIdx.x] = s2;
  __syncthreads();
  for (int off = TPB >> 1; off > 0; off >>= 1) {
    if (threadIdx.x < off) {
      r0[threadIdx.x] += r0[threadIdx.x + off];
      r1[threadIdx.x] += r1[threadIdx.x + off];
      r2[threadIdx.x] += r2[threadIdx.x + off];
    }
    __syncthreads();
  }
  if (threadIdx.x == 0) {
    Vp[4*b] = r0[0]; Vp[4*b+1] = r1[0]; Vp[4*b+2] = r2[0]; Vp[4*b+3] = 0.0f;
  }
}

__device__ __forceinline__ void wave_reduce_atomic(float local, float* dst, int lane) {
  for (int off = 16; off > 0; off >>= 1) local += __shfl_xor(local, off, 32);
  if (lane == 0) atomicAdd(dst, local);
}

// ---------------------------------------------------------------------------
// Shared tile machinery for the two fused logit+loss kernels.
// One wave per 16x16 output tile:
//   D = Q-tile(16x4) x V-tile^T(4x16) via V_WMMA_F32_16X16X4_F32.
// A-frag lane L: (M = L&15, K = 2*(L>>4)+{0,1}); B-frag symmetric, N = L&15.
// C/D layout: VGPR r, lane L -> M = 8*(L>>4)+r, N = L&15.
// Per-tile metadata (row ids, row bias, col ids, col bias) is loaded ONCE per
// lane and redistributed with shuffles computed BEFORE any divergence
// (bpermute from EXEC-disabled lanes returns 0).
// ---------------------------------------------------------------------------
struct TileCtx {
  v8f  c;          // logit tile (without biases)
  int   rid[8];    // gathered source-row index per output row
  float rb[8];     // row bias per output row
  int   gcol;      // gathered source-column index for this lane
  float cb;        // column bias for this lane
  int   half, l15;
  bool  interior;
};

__device__ __forceinline__ TileCtx make_tile(
    const float* __restrict__ Qp, const float* __restrict__ Vp,
    const int* __restrict__ rowIdx, const float* __restrict__ rowBias,
    const int* __restrict__ colIdx, const float* __restrict__ colBias,
    int i0, int rows, int j0, int cols, int lane)
{
  TileCtx t;
  t.half = lane >> 4; t.l15 = lane & 15;
  int arow = i0 + t.l15; if (arow > rows - 1) arow = rows - 1;
  int bcol = j0 + t.l15; if (bcol > cols - 1) bcol = cols - 1;
  v2f a = *(const v2f*)(Qp + 4*arow + 2*t.half);
  v2f b = *(const v2f*)(Vp + 4*bcol + 2*t.half);
  v8f c = {};
  t.c = __builtin_amdgcn_wmma_f32_16x16x4_f32(
      /*neg_a=*/false, a, /*neg_b=*/false, b,
      /*c_mod=*/(short)0, c, /*reuse_a=*/false, /*reuse_b=*/false);
  // lane (L&15) owns row i0+(L&15): source-row id + bias, loaded once
  int myRid   = rowIdx[arow];
  float myRb  = rowBias[myRid];
  t.gcol      = colIdx[bcol];
  t.cb        = colBias[t.gcol];
  // redistribute row metadata to the rows this lane accumulates (no divergence yet)
#pragma unroll
  for (int r = 0; r < 8; ++r) {
    int src = t.half * 8 + r;            // lanes 0..15 hold rows i0..i0+15
    t.rid[r] = __shfl(myRid, src, 32);
    t.rb[r]  = __shfl(myRb,  src, 32);
  }
  t.interior = (i0 + 16 <= rows) && (j0 + 16 <= cols);
  return t;
}

// ---------------------------------------------------------------------------
// K2: loss1 — WMMA logits tile + gathered-G squared error
// ---------------------------------------------------------------------------
__global__ void loss1_kernel(const float* __restrict__ G, const float* __restrict__ bg,
                             const float* __restrict__ b1,
                             const int* __restrict__ i1, const int* __restrict__ ig,
                             const float* __restrict__ Q1p, const float* __restrict__ Vp,
                             int m1, int mg, int gcols, int tilesN,
                             float* __restrict__ acc)
{
  int tile = blockIdx.x;
  int i0 = (tile / tilesN) * 16, j0 = (tile % tilesN) * 16;
  int lane = threadIdx.x;
  TileCtx t = make_tile(Q1p, Vp, i1, b1, ig, bg, i0, m1, j0, mg, lane);
  float local = 0.0f;
  if (t.interior) {
#pragma unroll
    for (int r = 0; r < 8; ++r) {
      float gv = G[(size_t)t.rid[r] * gcols + t.gcol];
      float d  = gv - (t.c[r] + t.cb + t.rb[r]);
      local += d * d;
    }
  } else {
#pragma unroll
    for (int r = 0; r < 8; ++r) {
      int gi = i0 + t.half*8 + r, gj = j0 + t.l15;
      if (gi < m1 && gj < mg) {
        float gv = G[(size_t)t.rid[r] * gcols + t.gcol];
        float d  = gv - (t.c[r] + t.cb + t.rb[r]);
        local += d * d;
      }
    }
  }
  wave_reduce_atomic(local, acc + 0, lane);
}

// ---------------------------------------------------------------------------
// K3: loss2 — WMMA logits tile + gathered-R stable BCE: softplus(z) - X*z
// ---------------------------------------------------------------------------
__global__ void loss2_kernel(const float* __restrict__ R, const float* __restrict__ br,
                             const float* __restrict__ b2,
                             const int* __restrict__ i2, const int* __restrict__ irr,
                             const float* __restrict__ Q2p, const float* __restrict__ Prp,
                             int m2, int mr, int rcols, int tilesN,
                             float* __restrict__ acc)
{
  int tile = blockIdx.x;
  int i0 = (tile / tilesN) * 16, j0 = (tile % tilesN) * 16;
  int lane = threadIdx.x;
  TileCtx t = make_tile(Q2p, Prp, i2, b2, irr, br, i0, m2, j0, mr, lane);
  float local = 0.0f;
  if (t.interior) {
#pragma unroll
    for (int r = 0; r < 8; ++r) {
      float X = R[(size_t)t.rid[r] * rcols + t.gcol];
      float z = t.c[r] + t.cb + t.rb[r];
      local += fmaxf(z, 0.0f) + log1pf(expf(-fabsf(z))) - X * z;
    }
  } else {
#pragma unroll
    for (int r = 0; r < 8; ++r) {
      int gi = i0 + t.half*8 + r, gj = j0 + t.l15;
      if (gi < m2 && gj < mr) {
        float X = R[(size_t)t.rid[r] * rcols + t.gcol];
        float z = t.c[r] + t.cb + t.rb[r];
        local += fmaxf(z, 0.0f) + log1pf(expf(-fabsf(z))) - X * z;
      }
    }
  }
  wave_reduce_atomic(local, acc + 1, lane);
}

// ---------------------------------------------------------------------------
// K4: MMD(P1, P2). Early-out when alpha[3]==0 (result multiplied by 0 anyway;
// loss4 is finite so alpha3*loss4 == 0 exactly). Fully computed otherwise.
// ---------------------------------------------------------------------------
__device__ __forceinline__ float gauss_multi(float d) {
  const float sig[19] = {1e-6f, 1e-5f, 1e-4f, 1e-3f, 1e-2f, 1e-1f, 1.f, 5.f, 10.f,
                         15.f, 20.f, 25.f, 30.f, 35.f, 100.f, 1e3f, 1e4f, 1e5f, 1e6f};
  float s = 0.0f;
#pragma unroll
  for (int k = 0; k < 19; ++k) s += expf(-(0.5f / sig[k]) * d);
  return s;
}

__global__ void mmd_kernel(const float* __restrict__ alpha,
                           const float* __restrict__ P1p, const float* __restrict__ P2p,
                           const float* __restrict__ nsq1, const float* __restrict__ nsq2,
                           int m1, int m2, float wxx, float wyy, float wxy,
                           float* __restrict__ acc)
{
  if (alpha[3] == 0.0f) return;   // uniform, data-dependent, deterministic
  __shared__ float red[TPB];
  int mm = m1 > m2 ? m1 : m2;
  long long t = (long long)blockIdx.x * blockDim.x + threadIdx.x;
  long long total = (long long)mm * mm;
  float term = 0.0f;
  if (t < total) {
    int i = (int)(t / mm), j = (int)(t % mm);
    float4 xi = {0,0,0,0}, yi = {0,0,0,0}, xj = {0,0,0,0}, yj = {0,0,0,0};
    if (i < m1) xi = *(const float4*)(P1p + 4*i);
    if (i < m2) yi = *(const float4*)(P2p + 4*i);
    if (j < m1) xj = *(const float4*)(P1p + 4*j);
    if (j < m2) yj = *(const float4*)(P2p + 4*j);
    if (i < m1 && j < m1) {
      float d = nsq1[i] + nsq1[j] - 2.0f*(xi.x*xj.x + xi.y*xj.y + xi.z*xj.z);
      term += wxx * gauss_multi(d);
    }
    if (i < m2 && j < m2) {
      float d = nsq2[i] + nsq2[j] - 2.0f*(yi.x*yj.x + yi.y*yj.y + yi.z*yj.z);
      term += wyy * gauss_multi(d);
    }
    if (i < m1 && j < m2) {
      float d = nsq1[i] + nsq2[j] - 2.0f*(xi.x*yj.x + xi.y*yj.y + xi.z*yj.z);
      term -= wxy * gauss_multi(d);
    }
  }
  red[threadIdx.x] = term;
  __syncthreads();
  for (int off = TPB >> 1; off > 0; off >>= 1) {
    if (threadIdx.x < off) red[threadIdx.x] += red[threadIdx.x + off];
    __syncthreads();
  }
  if (threadIdx.x == 0) atomicAdd(acc + 2, red[0]);
}

// ---------------------------------------------------------------------------
// K5: combine
// ---------------------------------------------------------------------------
__global__ void finalize_kernel(const float* __restrict__ alpha, const float* __restrict__ acc,
                                float inv1, float inv2, float* __restrict__ out)
{
  if (threadIdx.x == 0 && blockIdx.x == 0) {
    float l1 = acc[0] * inv1;
    float l2 = acc[1] * inv2;
    float l4 = acc[2];
    float t1 = alpha[0] * l1;
    float t2 = alpha[1] * l2;
    float t3 = alpha[2] * 0.0f;   // loss3 == 0 in mode 'C_12'
    float t4 = alpha[3] * l4;
    out[0] = t1 + t2 + t3 + t4;
    out[1] = t1; out[2] = t2; out[3] = t3; out[4] = t4;
  }
}

// ---------------------------------------------------------------------------
extern "C" void kernel_launch(void* const* d_in, const int* in_sizes, int n_in,
                              void* d_out, int out_size, void* d_ws, size_t ws_size,
                              hipStream_t stream) {
  const float* G   = (const float*)d_in[0];
  const float* R   = (const float*)d_in[1];
  const float* A   = (const float*)d_in[2];
  const float* C1  = (const float*)d_in[3];
  const float* C2  = (const float*)d_in[4];
  const float* Cr  = (const float*)d_in[5];
  const float* Ag  = (const float*)d_in[6];
  const float* Ar  = (const float*)d_in[7];
  const float* bg  = (const float*)d_in[8];
  const float* br  = (const float*)d_in[9];
  const float* b1  = (const float*)d_in[10];
  const float* b2  = (const float*)d_in[11];
  const float* alpha = (const float*)d_in[12];
  const int* i1  = (const int*)d_in[13];
  const int* i2  = (const int*)d_in[14];
  const int* igm = (const int*)d_in[15];
  const int* irm = (const int*)d_in[16];
  float* out = (float*)d_out;

  const int g  = in_sizes[8];    // b_g [1,g]
  const int rc = in_sizes[9];    // b_r [1,r]
  const int m1 = in_sizes[13];
  const int m2 = in_sizes[14];
  const int mg = in_sizes[15];
  const int mr = in_sizes[16];

  // workspace partition (floats, 16B aligned regions)
  float* ws = (float*)d_ws;
  size_t off = 0;
  auto take = [&](size_t nf) { float* p = ws + off; off += (nf + 3) & ~(size_t)3; return p; };
  float* S0   = take(rc);
  float* S1   = take(rc);
  float* S2   = take(rc);
  float* Prp  = take((size_t)4 * mr);
  float* P1p  = take((size_t)4 * m1);
  float* Q1p  = take((size_t)4 * m1);
  float* nsq1 = take(m1);
  float* P2p  = take((size_t)4 * m2);
  float* Q2p  = take((size_t)4 * m2);
  float* nsq2 = take(m2);
  float* Vp   = take((size_t)4 * mg);
  float* acc  = take(4);

  init_kernel<<<1, 32, 0, stream>>>(acc);

  int rmax = rc; if (mr > rmax) rmax = mr; if (m1 > rmax) rmax = m1; if (m2 > rmax) rmax = m2;
  prep_kernel<<<(rmax + TPB - 1) / TPB, TPB, 0, stream>>>(
      C1, C2, Cr, Ag, Ar, i1, i2, irm, rc, m1, m2, mr,
      S0, S1, S2, Prp, P1p, Q1p, nsq1, P2p, Q2p, nsq2);

  vrow_kernel<<<mg, TPB, 0, stream>>>(A, igm, S0, S1, S2, rc, mg, Vp);

  int tilesM1 = (m1 + 15) / 16, tilesN1 = (mg + 15) / 16;
  loss1_kernel<<<tilesM1 * tilesN1, 32, 0, stream>>>(
      G, bg, b1, i1, igm, Q1p, Vp, m1, mg, g, tilesN1, acc);

  int tilesM2 = (m2 + 15) / 16, tilesN2 = (mr + 15) / 16;
  loss2_kernel<<<tilesM2 * tilesN2, 32, 0, stream>>>(
      R, br, b2, i2, irm, Q2p, Prp, m2, mr, rc, tilesN2, acc);

  int mm = m1 > m2 ? m1 : m2;
  long long total = (long long)mm * mm;
  int mblocks = (int)((total + TPB - 1) / TPB);
  mmd_kernel<<<mblocks, TPB, 0, stream>>>(
      alpha, P1p, P2p, nsq1, nsq2, m1, m2,
      1.0f / ((float)m1 * (float)m1), 1.0f / ((float)m2 * (float)m2),
      2.0f / ((float)m1 * (float)m2), acc);

  finalize_kernel<<<1, 32, 0, stream>>>(
      alpha, acc, 1.0f / ((float)m1 * (float)mg), 1.0f / ((float)m2 * (float)mr), out);
}